// CTCGreedyDecoder_54073638257107
// MI455X (gfx1250) — compile-verified
//
#include <hip/hip_runtime.h>
#include <hip/hip_bf16.h>

#define B_ 1024
#define T_ 256
#define V_ 64
#define S_ 513          // 2*T + 1
#define BLANK_ 63
#define NEGV (-1.0e30f)

// One workgroup (512 threads = 16 waves) per batch row.
__global__ __launch_bounds__(512) void ctc_greedy_loss_kernel(
    const float* __restrict__ logits,   // [B, T, V]
    float* __restrict__ out)            // labels[B*T] | lengths[B] | prob[B]
{
    const int b   = blockIdx.x;
    const int tid = threadIdx.x;

    __shared__ int   pred_s[T_];
    __shared__ float lse_s[T_];
    __shared__ int   labels_s[T_];
    __shared__ int   len_s;
    __shared__ float alpha_s[2][S_ + 7];
    __shared__ __align__(16) float row_s[2][V_];

    const float* base = logits + (size_t)b * T_ * V_;

    // ---- async prefetch of DP row 0 into row_s[0] (wave 0: 32 lanes x 8B = 256B) ----
    if (tid < 32) {
        unsigned long long ga = (unsigned long long)(size_t)base + (unsigned long long)(tid * 8);
        unsigned ld = (unsigned)(size_t)(void*)(&row_s[0][0]) + (unsigned)(tid * 8);
        asm volatile("global_load_async_to_lds_b64 %0, %1, off"
                     :: "v"(ld), "v"(ga) : "memory");
    }

    if (tid < T_) labels_s[tid] = BLANK_;

    // ---- Phase A: per-timestep argmax + logsumexp (thread == timestep) ----
    if (tid < T_) {
        const float4* p = (const float4*)(base + (size_t)tid * V_);
        float4 r[16];
        #pragma unroll
        for (int j = 0; j < 16; ++j) r[j] = p[j];

        float best = -3.4e38f; int bi = 0;
        #pragma unroll
        for (int j = 0; j < 16; ++j) {
            if (r[j].x > best) { best = r[j].x; bi = 4*j + 0; }
            if (r[j].y > best) { best = r[j].y; bi = 4*j + 1; }
            if (r[j].z > best) { best = r[j].z; bi = 4*j + 2; }
            if (r[j].w > best) { best = r[j].w; bi = 4*j + 3; }
        }
        const float mx = best;   // row max (argmax value)
        float sum = 0.f;
        #pragma unroll
        for (int j = 0; j < 16; ++j) {
            sum += __expf(r[j].x - mx) + __expf(r[j].y - mx)
                 + __expf(r[j].z - mx) + __expf(r[j].w - mx);
        }
        pred_s[tid] = bi;
        lse_s[tid]  = mx + __logf(sum);
    }
    __syncthreads();

    // ---- Phase B: merge repeats, drop blanks (tiny sequential scan) ----
    if (tid == 0) {
        int prev = -1, len = 0;
        for (int t = 0; t < T_; ++t) {
            int p = pred_s[t];
            if (p != BLANK_ && p != prev) labels_s[len++] = p;
            prev = p;
        }
        len_s = len;
    }
    __syncthreads();

    // decoded labels out (dense, blank-padded), as float
    if (tid < T_) out[(size_t)b * T_ + tid] = (float)labels_s[tid];

    // init alpha[0]
    for (int s = tid; s < S_; s += 512) alpha_s[0][s] = (s == 0) ? 0.f : NEGV;

    // per-state invariants (state s0 = tid; thread 0 also owns s = 512)
    const int  s0 = tid;
    const int  c0 = (s0 & 1) ? labels_s[s0 >> 1] : BLANK_;
    const bool k0 = (s0 & 1) && (s0 >= 3) && (c0 != BLANK_) &&
                    (c0 != labels_s[(s0 >> 1) - 1]);

    if (tid < 32) asm volatile("s_wait_asynccnt 0" ::: "memory");
    __syncthreads();

    // ---- Phase C: forward DP over T steps, async-streamed logit rows ----
    for (int t = 0; t < T_; ++t) {
        const int pr = t & 1, nx = pr ^ 1;

        // prefetch row t+1 into the other LDS row buffer
        if (tid < 32 && (t + 1) < T_) {
            unsigned long long ga = (unsigned long long)(size_t)(base + (size_t)(t + 1) * V_)
                                  + (unsigned long long)(tid * 8);
            unsigned ld = (unsigned)(size_t)(void*)(&row_s[nx][0]) + (unsigned)(tid * 8);
            asm volatile("global_load_async_to_lds_b64 %0, %1, off"
                         :: "v"(ld), "v"(ga) : "memory");
        }

        const float  lseT = lse_s[t];
        const float* aOld = alpha_s[pr];
        float*       aNew = alpha_s[nx];

        {   // state s0 = tid  (0..511)
            const int s = s0;
            float a1 = aOld[s];
            float a2 = (s >= 1) ? aOld[s - 1] : NEGV;
            float a3 = k0 ? aOld[s - 2] : NEGV;
            float m  = fmaxf(a1, fmaxf(a2, a3));
            float sm = __expf(a1 - m) + __expf(a2 - m) + __expf(a3 - m);
            aNew[s]  = m + __logf(sm) + (row_s[pr][c0] - lseT);
        }
        if (tid == 0) {   // state s = 512 (even -> blank, no skip transition)
            float a1 = aOld[512];
            float a2 = aOld[511];
            float m  = fmaxf(a1, a2);
            float sm = __expf(a1 - m) + __expf(a2 - m);
            aNew[512] = m + __logf(sm) + (row_s[pr][BLANK_] - lseT);
        }

        if (tid < 32) asm volatile("s_wait_asynccnt 0" ::: "memory");
        __syncthreads();
    }

    // ---- Epilogue: loss -> probability ----
    if (tid == 0) {
        const float* aF = alpha_s[T_ & 1];
        const int len = len_s;
        const int end = 2 * len;
        float e1 = aF[end];
        float e2 = (len > 0) ? aF[end - 1] : NEGV;
        float m  = fmaxf(e1, e2);
        float ll = m + __logf(__expf(e1 - m) + __expf(e2 - m));  // = -loss
        out[(size_t)B_ * T_ + b]       = (float)len;   // lengths
        out[(size_t)B_ * T_ + B_ + b]  = __expf(ll);   // probability
    }
}

extern "C" void kernel_launch(void* const* d_in, const int* in_sizes, int n_in,
                              void* d_out, int out_size, void* d_ws, size_t ws_size,
                              hipStream_t stream) {
    const float* logits = (const float*)d_in[0];
    float* out = (float*)d_out;
    (void)in_sizes; (void)n_in; (void)out_size; (void)d_ws; (void)ws_size;
    hipLaunchKernelGGL(ctc_greedy_loss_kernel, dim3(B_), dim3(512), 0, stream,
                       logits, out);
}